// GATSBY_59047210385934
// MI455X (gfx1250) — compile-verified
//
#include <hip/hip_runtime.h>
#include <math.h>

typedef __attribute__((ext_vector_type(2))) float v2f;
typedef __attribute__((ext_vector_type(8))) float v8f;

#define GAT_NEG_SLOPE 0.2f

__device__ __forceinline__ float gat_elu(float x) {
  return x > 0.0f ? x : (expf(x) - 1.0f);
}
__device__ __forceinline__ float gat_leaky(float x) {
  return x > 0.0f ? x : GAT_NEG_SLOPE * x;
}
// float atomic max via sign-split: one hardware atomic, no CAS loop.
__device__ __forceinline__ void gat_atomic_max_f32(float* addr, float val) {
  if (!(val < 0.0f)) {
    atomicMax((int*)addr, __float_as_int(val));
  } else {
    atomicMin((unsigned int*)addr, (unsigned int)__float_as_int(val));
  }
}
// edges 0..E-1 come from edge_index rows; edges E..E+N-1 are the self-loops
__device__ __forceinline__ void gat_edge_nodes(const int* __restrict__ ei, int E,
                                               int e, int& s, int& d) {
  if (e < E) { s = ei[e]; d = ei[E + e]; }
  else       { s = e - E; d = s; }
}

// ---------------------------------------------------------------------------
// fp32 WMMA GEMM: C[M,N] = A[M,K] @ B[K,N] (+bias), row-major.
// One wave -> 16x32 output (two 16x16 accumulators sharing the A fragment),
// block = 4 waves -> 16(M) x 128(N). V_WMMA_F32_16X16X4_F32, K stepped by 4.
// Out-of-range columns are handled WITHOUT exec divergence: B column index is
// clamped to a legal address (garbage feeds only dead output columns) and the
// store is guarded. M%16==0 and K%4==0 hold for all three GEMMs.
// ---------------------------------------------------------------------------
__global__ __launch_bounds__(128)
void gat_gemm_f32_wmma(const float* __restrict__ A, const float* __restrict__ B,
                       const float* __restrict__ bias, float* __restrict__ C,
                       int M, int K, int Nn)
{
  const int lane = threadIdx.x & 31;
  const int wave = threadIdx.x >> 5;
  const int m0   = blockIdx.x * 16;
  const int n0   = blockIdx.y * 128 + wave * 32;
  const int half = lane >> 4;        // 0: K={0,1}, 1: K={2,3} of each step
  const int l16  = lane & 15;
  const int kb   = half * 2;
  const int aM   = m0 + l16;         // A fragment: lanes 0-15 carry rows
  const int bN0  = n0 + l16;         // first  16-col tile
  const int bN1  = n0 + 16 + l16;    // second 16-col tile
  const int bC0  = (bN0 < Nn) ? bN0 : (Nn - 1);  // clamped legal column
  const int bC1  = (bN1 < Nn) ? bN1 : (Nn - 1);

  const float* __restrict__ Ap  = A + (size_t)aM * K + kb;
  const float* __restrict__ Bp0 = B + (size_t)kb * Nn + bC0;
  const float* __restrict__ Bp1 = B + (size_t)kb * Nn + bC1;
  const size_t bStep = (size_t)4 * Nn;

  v8f acc0 = {0.f, 0.f, 0.f, 0.f, 0.f, 0.f, 0.f, 0.f};
  v8f acc1 = {0.f, 0.f, 0.f, 0.f, 0.f, 0.f, 0.f, 0.f};

  #pragma unroll 2
  for (int k = 0; k < K; k += 4) {
    v2f a, b0, b1;
    a.x  = Ap[0];
    a.y  = Ap[1];
    b0.x = Bp0[0];
    b0.y = Bp0[Nn];
    b1.x = Bp1[0];
    b1.y = Bp1[Nn];
    acc0 = __builtin_amdgcn_wmma_f32_16x16x4_f32(false, a, false, b0,
                                                 (short)0, acc0, false, false);
    acc1 = __builtin_amdgcn_wmma_f32_16x16x4_f32(false, a, false, b1,
                                                 (short)0, acc1, false, false);
    Ap  += 4;
    Bp0 += bStep;
    Bp1 += bStep;
  }

  const int rowBase = m0 + half * 8;   // C/D layout: VGPR r -> row rowBase+r
  if (bN0 < Nn) {
    const float ba = bias ? bias[bN0] : 0.0f;
    #pragma unroll
    for (int r = 0; r < 8; ++r)
      C[(size_t)(rowBase + r) * Nn + bN0] = acc0[r] + ba;
  }
  if (bN1 < Nn) {
    const float ba = bias ? bias[bN1] : 0.0f;
    #pragma unroll
    for (int r = 0; r < 8; ++r)
      C[(size_t)(rowBase + r) * Nn + bN1] = acc1[r] + ba;
  }
}

// a_src[n,h] = <h[n,h,:], att_src[h,:]>, a_dst likewise. h viewed as [N*H, 64].
__global__ __launch_bounds__(256)
void gat_att_dot(const float* __restrict__ h, const float* __restrict__ asrc,
                 const float* __restrict__ adst, float* __restrict__ as_,
                 float* __restrict__ ad_, int NH, int hmask)
{
  int t = blockIdx.x * blockDim.x + threadIdx.x;
  if (t >= NH) return;
  int hh = t & hmask;
  const float* __restrict__ hp = h + (size_t)t * 64;
  const float* __restrict__ sp = asrc + hh * 64;
  const float* __restrict__ dp = adst + hh * 64;
  float s = 0.f, d = 0.f;
  #pragma unroll 8
  for (int c = 0; c < 64; ++c) { float v = hp[c]; s += v * sp[c]; d += v * dp[c]; }
  as_[t] = s;
  ad_[t] = d;
}

__global__ __launch_bounds__(256)
void gat_init_softmax(float* __restrict__ m, float* __restrict__ den, int n)
{
  int t = blockIdx.x * blockDim.x + threadIdx.x;
  if (t < n) { m[t] = -INFINITY; den[t] = 0.f; }
}

// pass 1: segment max of leaky(a_s[src]+a_d[dst]) over incoming edges of dst
__global__ __launch_bounds__(256)
void gat_edge_max(const int* __restrict__ ei, int E, int total, int hmask, int hshift,
                  const float* __restrict__ as_, const float* __restrict__ ad_,
                  float* __restrict__ m)
{
  int t = blockIdx.x * blockDim.x + threadIdx.x;
  if (t >= total) return;
  int hh = t & hmask;
  int e  = t >> hshift;
  int H  = hmask + 1;
  int s, d; gat_edge_nodes(ei, E, e, s, d);
  float v = gat_leaky(as_[s * H + hh] + ad_[d * H + hh]);
  gat_atomic_max_f32(&m[d * H + hh], v);
}

// pass 2: denom[dst] += exp(e - m[dst])
__global__ __launch_bounds__(256)
void gat_edge_sum(const int* __restrict__ ei, int E, int total, int hmask, int hshift,
                  const float* __restrict__ as_, const float* __restrict__ ad_,
                  const float* __restrict__ m, float* __restrict__ den)
{
  int t = blockIdx.x * blockDim.x + threadIdx.x;
  if (t >= total) return;
  int hh = t & hmask;
  int e  = t >> hshift;
  int H  = hmask + 1;
  int s, d; gat_edge_nodes(ei, E, e, s, d);
  float v = gat_leaky(as_[s * H + hh] + ad_[d * H + hh]);
  atomicAdd(&den[d * H + hh], expf(v - m[d * H + hh]));
}

// pass 3: out[dst,h,c] += h[src,h,c] * alpha(e,h). One thread per (edge,h,c);
// c = lane%64 -> fully coalesced 64-wide gather + atomic scatter.
__global__ __launch_bounds__(256)
void gat_edge_agg(const int* __restrict__ ei, int E, long long total,
                  int hmask, int hshift,
                  const float* __restrict__ as_, const float* __restrict__ ad_,
                  const float* __restrict__ m, const float* __restrict__ den,
                  const float* __restrict__ hf, float* __restrict__ outp)
{
  long long t = (long long)blockIdx.x * blockDim.x + threadIdx.x;
  if (t >= total) return;
  int c = (int)(t & 63);
  long long r = t >> 6;             // (edge, head) pair
  int hh = (int)(r & hmask);
  int e  = (int)(r >> hshift);
  int H  = hmask + 1;
  int s, d; gat_edge_nodes(ei, E, e, s, d);
  float v = gat_leaky(as_[s * H + hh] + ad_[d * H + hh]);
  float alpha = expf(v - m[d * H + hh]) / (den[d * H + hh] + 1e-16f);
  atomicAdd(&outp[((size_t)d * H + hh) * 64 + c],
            hf[((size_t)s * H + hh) * 64 + c] * alpha);
}

__global__ __launch_bounds__(256)
void gat_bias_elu(float* __restrict__ x, const float* __restrict__ b,
                  int total, int dmask)
{
  int t = blockIdx.x * blockDim.x + threadIdx.x;
  if (t < total) x[t] = gat_elu(x[t] + b[t & dmask]);
}

extern "C" void kernel_launch(void* const* d_in, const int* in_sizes, int n_in,
                              void* d_out, int out_size, void* d_ws, size_t ws_size,
                              hipStream_t stream) {
  const float* x        = (const float*)d_in[0];
  const int*   ei       = (const int*)d_in[1];
  const float* W1       = (const float*)d_in[2];
  const float* att_src1 = (const float*)d_in[3];
  const float* att_dst1 = (const float*)d_in[4];
  const float* b1       = (const float*)d_in[5];
  const float* W2       = (const float*)d_in[6];
  const float* att_src2 = (const float*)d_in[7];
  const float* att_dst2 = (const float*)d_in[8];
  const float* b2       = (const float*)d_in[9];
  const float* Wl       = (const float*)d_in[10];
  const float* bl       = (const float*)d_in[11];
  float* out = (float*)d_out;

  const int F = 500, D1 = 512, H1 = 8, C2 = 64;
  const int N    = in_sizes[0] / F;     // 50000
  const int E    = in_sizes[1] / 2;     // 600000
  const int Etot = E + N;               // with self-loops

  // workspace layout (fp32)
  float* ws = (float*)d_ws;
  size_t o = 0;
  float* h1   = ws + o; o += (size_t)N * D1;   // x@W1              [N,8,64]
  float* out1 = ws + o; o += (size_t)N * D1;   // layer-1 aggregate [N,8,64]
  float* as1  = ws + o; o += (size_t)N * H1;
  float* ad1  = ws + o; o += (size_t)N * H1;
  float* m1   = ws + o; o += (size_t)N * H1;
  float* den1 = ws + o; o += (size_t)N * H1;
  float* h2   = ws + o; o += (size_t)N * C2;   // act1@W2           [N,64]
  float* out2 = ws + o; o += (size_t)N * C2;   // layer-2 aggregate [N,64]
  float* as2  = ws + o; o += (size_t)N;
  float* ad2  = ws + o; o += (size_t)N;
  float* m2   = ws + o; o += (size_t)N;
  float* den2 = ws + o; o += (size_t)N;

  const int B = 256;

  // ---- layer 1 ----
  gat_gemm_f32_wmma<<<dim3(N / 16, D1 / 128), 128, 0, stream>>>(
      x, W1, nullptr, h1, N, F, D1);

  const int NH1 = N * H1;
  gat_att_dot<<<(NH1 + B - 1) / B, B, 0, stream>>>(
      h1, att_src1, att_dst1, as1, ad1, NH1, 7);

  gat_init_softmax<<<(NH1 + B - 1) / B, B, 0, stream>>>(m1, den1, NH1);
  hipMemsetAsync(out1, 0, (size_t)N * D1 * sizeof(float), stream);

  const int T1 = Etot * H1;
  gat_edge_max<<<(T1 + B - 1) / B, B, 0, stream>>>(ei, E, T1, 7, 3, as1, ad1, m1);
  gat_edge_sum<<<(T1 + B - 1) / B, B, 0, stream>>>(ei, E, T1, 7, 3, as1, ad1, m1, den1);

  const long long TA1 = (long long)Etot * H1 * 64;
  gat_edge_agg<<<(unsigned)((TA1 + B - 1) / B), B, 0, stream>>>(
      ei, E, TA1, 7, 3, as1, ad1, m1, den1, h1, out1);

  gat_bias_elu<<<((N * D1) + B - 1) / B, B, 0, stream>>>(out1, b1, N * D1, D1 - 1);

  // ---- layer 2 (single head) ----
  gat_gemm_f32_wmma<<<dim3(N / 16, 1), 128, 0, stream>>>(
      out1, W2, nullptr, h2, N, D1, C2);

  gat_att_dot<<<(N + B - 1) / B, B, 0, stream>>>(
      h2, att_src2, att_dst2, as2, ad2, N, 0);

  gat_init_softmax<<<(N + B - 1) / B, B, 0, stream>>>(m2, den2, N);
  hipMemsetAsync(out2, 0, (size_t)N * C2 * sizeof(float), stream);

  gat_edge_max<<<(Etot + B - 1) / B, B, 0, stream>>>(ei, E, Etot, 0, 0, as2, ad2, m2);
  gat_edge_sum<<<(Etot + B - 1) / B, B, 0, stream>>>(ei, E, Etot, 0, 0, as2, ad2, m2, den2);

  const long long TA2 = (long long)Etot * 64;
  gat_edge_agg<<<(unsigned)((TA2 + B - 1) / B), B, 0, stream>>>(
      ei, E, TA2, 0, 0, as2, ad2, m2, den2, h2, out2);

  gat_bias_elu<<<((N * C2) + B - 1) / B, B, 0, stream>>>(out2, b2, N * C2, C2 - 1);

  // ---- linear head: out = act2 @ Wl + bl  (N=500 -> column tail guarded) ----
  gat_gemm_f32_wmma<<<dim3(N / 16, (F + 127) / 128), 128, 0, stream>>>(
      out2, Wl, bl, out, N, C2, F);
}